// KLDivergenceLoss_87058987090230
// MI455X (gfx1250) — compile-verified
//
#include <hip/hip_runtime.h>

// NLL gather + mean:  out = mean_i( -pred[i, target[i]] ),  N=4096, C=50257.
// Latency-bound: one gathered load per thread, WMMA-based f32 tree reduction.

typedef __attribute__((ext_vector_type(2))) float v2f;
typedef __attribute__((ext_vector_type(8))) float v8f;

#define BLOCK 128  // 4 waves per block (wave32)

// Reduce 64 f32 values (2 VGPRs across the wave) into C via
// D = A x ones(4x16) + C  ==>  D[m][n] = rowsum(A)[m] + C[m][n].
__device__ __forceinline__ v8f wmma_accum64(v2f a, v8f c) {
    v2f ones = {1.0f, 1.0f};  // entire 4x16 B matrix = 1.0
    return __builtin_amdgcn_wmma_f32_16x16x4_f32(
        /*neg_a=*/false, a, /*neg_b=*/false, ones,
        /*c_mod=*/(short)0, c, /*reuse_a=*/false, /*reuse_b=*/false);
}

// After the chained WMMAs, D columns are identical (B was all-ones):
// lanes 0-15 hold D[0..7][n] in c[0..7]; lanes 16-31 hold D[8..15][n].
// Sum the 8 in-lane elements, then fold the two half-waves together.
__device__ __forceinline__ float wmma_collapse(v8f c) {
    float s = c[0] + c[1] + c[2] + c[3] + c[4] + c[5] + c[6] + c[7];
    s += __shfl_xor(s, 16, 32);
    return s;  // full-wave total in every lane
}

__global__ void __launch_bounds__(BLOCK)
nll_partial_kernel(const float* __restrict__ pred,
                   const int* __restrict__ tgt,
                   float* __restrict__ partials,
                   int N, int C) {
    __shared__ float sm[BLOCK];
    int i = blockIdx.x * BLOCK + threadIdx.x;
    float p = 0.0f;
    if (i < N) {
        int t = tgt[i];
        p = -pred[(long long)i * (long long)C + (long long)t];
    }
    sm[threadIdx.x] = p;
    __syncthreads();

    // Wave 0 (threads 0-31 exactly, wave32) reduces all 128 partials.
    // EXEC is all-ones within this wave, as WMMA requires.
    if (threadIdx.x < 32) {
        int lane = threadIdx.x;
        v8f c = {};
        #pragma unroll
        for (int chunk = 0; chunk < BLOCK / 64; ++chunk) {
            v2f a;
            a.x = sm[chunk * 64 + lane];
            a.y = sm[chunk * 64 + 32 + lane];
            c = wmma_accum64(a, c);
        }
        float s = wmma_collapse(c);
        if (lane == 0) partials[blockIdx.x] = s;
    }
}

__global__ void __launch_bounds__(32)
nll_final_kernel(const float* __restrict__ partials,
                 float* __restrict__ out,
                 int nparts, float invN) {
    int lane = threadIdx.x;  // single wave, EXEC all-ones
    float p = 0.0f;
    for (int j = lane; j < nparts; j += 32) p += partials[j];
    v2f a;
    a.x = p;
    a.y = 0.0f;
    v8f c = {};
    c = wmma_accum64(a, c);
    float s = wmma_collapse(c);
    if (lane == 0) out[0] = s * invN;
}

extern "C" void kernel_launch(void* const* d_in, const int* in_sizes, int n_in,
                              void* d_out, int out_size, void* d_ws, size_t ws_size,
                              hipStream_t stream) {
    const float* pred = (const float*)d_in[0];   // (N, C) float32 log-probs
    const int* tgt    = (const int*)d_in[1];     // (N,) int
    float* out        = (float*)d_out;           // scalar float32
    float* partials   = (float*)d_ws;            // >= nblocks floats

    const int N = in_sizes[1];                   // 4096
    const int C = in_sizes[0] / N;               // 50257
    const int nblocks = (N + BLOCK - 1) / BLOCK; // 32

    nll_partial_kernel<<<nblocks, BLOCK, 0, stream>>>(pred, tgt, partials, N, C);
    nll_final_kernel<<<1, 32, 0, stream>>>(partials, out, nblocks, 1.0f / (float)N);
}